// GSAGE_51187420233864
// MI455X (gfx1250) — compile-verified
//
#include <hip/hip_runtime.h>

typedef __attribute__((ext_vector_type(2))) float v2f;
typedef __attribute__((ext_vector_type(8))) float v8f;

#define INC 64
#define HIDC 128

// ---------------------------------------------------------------- zero ws
__global__ void gs_zero(float4* __restrict__ p, long long n4) {
  long long i = (long long)blockIdx.x * blockDim.x + threadIdx.x;
  long long stride = (long long)gridDim.x * blockDim.x;
  const float4 z = {0.f, 0.f, 0.f, 0.f};
  for (; i < n4; i += stride) p[i] = z;
}

// ------------------------------------------------- edge scatter (segment sum)
// C/4 lanes cover one edge's feature row with float4 gathers (global_load_b128)
// followed by 4 hardware f32 atomics per lane into sum[dst]. Coalesced on both.
template <int C, int COUNT_DEG>
__global__ void gs_scatter(const float* __restrict__ x,
                           const int* __restrict__ src,
                           const int* __restrict__ dst,
                           float* __restrict__ sum,
                           float* __restrict__ deg,
                           int nEdges) {
  const int tpe = C / 4;                 // threads per edge
  const int edgesPerBlock = 256 / tpe;
  int eLocal = threadIdx.x / tpe;
  int c4 = (threadIdx.x % tpe) * 4;
  int e = blockIdx.x * edgesPerBlock + eLocal;
  if (e >= nEdges) return;
  int s = src[e];
  int d = dst[e];
  const float4 v = *(const float4*)&x[(long long)s * C + c4];
  float* o = &sum[(long long)d * C + c4];
  unsafeAtomicAdd(o + 0, v.x);           // global_atomic_add_f32
  unsafeAtomicAdd(o + 1, v.y);
  unsafeAtomicAdd(o + 2, v.z);
  unsafeAtomicAdd(o + 3, v.w);
  if (COUNT_DEG && c4 == 0) unsafeAtomicAdd(&deg[d], 1.0f);
}

// ------------------------------------------------- sum -> mean (in place)
template <int C>
__global__ void gs_mean(float4* __restrict__ sum, const float* __restrict__ deg,
                        long long total4) {
  long long i = (long long)blockIdx.x * blockDim.x + threadIdx.x;
  if (i >= total4) return;
  float r = 1.0f / fmaxf(deg[(i * 4) / C], 1.0f);
  float4 v = sum[i];
  v.x *= r; v.y *= r; v.z *= r; v.w *= r;
  sum[i] = v;
}

// ------------------------------------------------- layer 1 GEMM via WMMA f32
// h[m, n] = relu( mean1[m,:]@W1l[n,:] + x[m,:]@W1r[n,:] + b1[n] ),  K = 64+64.
// One wave per 16x16 output tile; 32 x V_WMMA_F32_16X16X4_F32 (two split halves
// so each half has constant-stride addressing, no per-step operand selects).
__global__ __launch_bounds__(256) void gs_gemm1(
    const float* __restrict__ mean1, const float* __restrict__ x,
    const float* __restrict__ W1l, const float* __restrict__ W1r,
    const float* __restrict__ b1, float* __restrict__ h, int nNodes) {
  const int lane   = threadIdx.x;            // 0..31 (wave32)
  const int tile_n = threadIdx.y;            // 0..7
  const int tile_m = blockIdx.x;
  const int m  = tile_m * 16 + (lane & 15);  // A row this lane feeds
  const int n  = tile_n * 16 + (lane & 15);  // B col this lane feeds
  const int kh = (lane >> 4) * 2;            // K sub-slot {0,1} vs {2,3}
  const float* aP0 = mean1 + (long long)m * INC + kh;  // K half 1 (neighbor mean)
  const float* aP1 = x     + (long long)m * INC + kh;  // K half 2 (root)
  const float* bP0 = W1l + (long long)n * INC + kh;
  const float* bP1 = W1r + (long long)n * INC + kh;

  v8f acc = {};
  #pragma unroll 4
  for (int kt = 0; kt < INC / 4; ++kt) {     // K = 0..63 : mean1 x W1l
    v2f a, b;
    a.x = aP0[kt * 4]; a.y = aP0[kt * 4 + 1];
    b.x = bP0[kt * 4]; b.y = bP0[kt * 4 + 1];
    acc = __builtin_amdgcn_wmma_f32_16x16x4_f32(
        false, a, false, b, (short)0, acc, false, false);
  }
  #pragma unroll 4
  for (int kt = 0; kt < INC / 4; ++kt) {     // K = 64..127 : x x W1r
    v2f a, b;
    a.x = aP1[kt * 4]; a.y = aP1[kt * 4 + 1];
    b.x = bP1[kt * 4]; b.y = bP1[kt * 4 + 1];
    acc = __builtin_amdgcn_wmma_f32_16x16x4_f32(
        false, a, false, b, (short)0, acc, false, false);
  }

  const float bias = b1[n];
  const int rbase = tile_m * 16 + ((lane >> 4) ? 8 : 0);   // C/D row layout
  #pragma unroll
  for (int r = 0; r < 8; ++r) {
    float v = fmaxf(acc[r] + bias, 0.0f);                  // relu
    h[(long long)(rbase + r) * HIDC + n] = v;
  }
}

// ------------------------------------------------- heads GEMM via WMMA f32
// threadIdx.y==0: y_class [N,16]; ==1: y_domain [N,2]. K = 128+128.
__global__ __launch_bounds__(64) void gs_gemm2(
    const float* __restrict__ mean2, const float* __restrict__ h,
    const float* __restrict__ W2l, const float* __restrict__ W2r,
    const float* __restrict__ b2,
    const float* __restrict__ Wdl, const float* __restrict__ Wdr,
    const float* __restrict__ bd,
    float* __restrict__ out, int nNodes) {
  const int lane = threadIdx.x;
  const int head = threadIdx.y;              // 0 = class, 1 = domain
  const int tile_m = blockIdx.x;
  const int n  = lane & 15;
  const int m  = tile_m * 16 + n;
  const int kh = (lane >> 4) * 2;
  const int nOut = head ? 2 : 16;
  // Branchless masked weight row: clamp OOB lanes to row 0, scale loads by 0.
  const float msk = (n < nOut) ? 1.0f : 0.0f;
  const int nw = (n < nOut) ? n : 0;
  const float* Wl = head ? Wdl : W2l;
  const float* Wr = head ? Wdr : W2r;
  const float* aP0 = mean2 + (long long)m * HIDC + kh;
  const float* aP1 = h     + (long long)m * HIDC + kh;
  const float* bP0 = Wl + (long long)nw * HIDC + kh;
  const float* bP1 = Wr + (long long)nw * HIDC + kh;

  v8f acc = {};
  #pragma unroll 4
  for (int kt = 0; kt < HIDC / 4; ++kt) {    // K = 0..127 : mean2 x Wl
    v2f a, b;
    a.x = aP0[kt * 4]; a.y = aP0[kt * 4 + 1];
    b.x = bP0[kt * 4] * msk; b.y = bP0[kt * 4 + 1] * msk;
    acc = __builtin_amdgcn_wmma_f32_16x16x4_f32(
        false, a, false, b, (short)0, acc, false, false);
  }
  #pragma unroll 4
  for (int kt = 0; kt < HIDC / 4; ++kt) {    // K = 128..255 : h x Wr
    v2f a, b;
    a.x = aP1[kt * 4]; a.y = aP1[kt * 4 + 1];
    b.x = bP1[kt * 4] * msk; b.y = bP1[kt * 4 + 1] * msk;
    acc = __builtin_amdgcn_wmma_f32_16x16x4_f32(
        false, a, false, b, (short)0, acc, false, false);
  }

  const int rbase = tile_m * 16 + ((lane >> 4) ? 8 : 0);
  if (head == 0) {
    const float bias = b2[n];
    #pragma unroll
    for (int r = 0; r < 8; ++r)
      out[(long long)(rbase + r) * 16 + n] = acc[r] + bias;
  } else if (n < 2) {
    const float bias = bd[n];
    const long long domBase = (long long)nNodes * 16;
    #pragma unroll
    for (int r = 0; r < 8; ++r)
      out[domBase + (long long)(rbase + r) * 2 + n] = acc[r] + bias;
  }
}

// =============================================================== launcher
extern "C" void kernel_launch(void* const* d_in, const int* in_sizes, int n_in,
                              void* d_out, int out_size, void* d_ws, size_t ws_size,
                              hipStream_t stream) {
  const float* x    = (const float*)d_in[0];
  const int*   edge = (const int*)d_in[1];     // [2, E]
  const float* W1l  = (const float*)d_in[2];
  const float* W1r  = (const float*)d_in[3];
  const float* b1   = (const float*)d_in[4];
  const float* W2l  = (const float*)d_in[5];
  const float* W2r  = (const float*)d_in[6];
  const float* b2   = (const float*)d_in[7];
  const float* Wdl  = (const float*)d_in[8];
  const float* Wdr  = (const float*)d_in[9];
  const float* bd   = (const float*)d_in[10];
  float* out = (float*)d_out;

  const int nNodes = in_sizes[0] / INC;
  const int nEdges = in_sizes[1] / 2;
  const int* src = edge;
  const int* dst = edge + nEdges;

  // workspace layout (floats): deg[N] | sum1/mean1[N*64] | sum2/mean2[N*128] | h[N*128]
  float* ws    = (float*)d_ws;
  float* deg   = ws;
  float* sum1  = deg + nNodes;
  float* sum2  = sum1 + (long long)nNodes * INC;
  float* h     = sum2 + (long long)nNodes * HIDC;
  const long long zeroCount = (long long)nNodes * (1 + INC + HIDC); // deg+sum1+sum2

  // 1) zero accumulators (b128 stores; counts here are multiples of 4)
  gs_zero<<<2048, 256, 0, stream>>>((float4*)ws, zeroCount / 4);

  // 2) segment-sum of x into sum1, degree count (16 lanes/edge, float4 gathers)
  {
    int blocks = (nEdges + 15) / 16;
    gs_scatter<INC, 1><<<blocks, 256, 0, stream>>>(x, src, dst, sum1, deg, nEdges);
  }
  // 3) sum1 -> mean1
  {
    long long total4 = (long long)nNodes * INC / 4;
    gs_mean<INC><<<(int)((total4 + 255) / 256), 256, 0, stream>>>(
        (float4*)sum1, deg, total4);
  }
  // 4) h = relu(mean1@W1l^T + x@W1r^T + b1) via WMMA f32
  gs_gemm1<<<dim3(nNodes / 16), dim3(32, 8), 0, stream>>>(sum1, x, W1l, W1r, b1, h, nNodes);

  // 5) segment-sum of h into sum2 (32 lanes/edge)
  {
    int blocks = (nEdges + 7) / 8;
    gs_scatter<HIDC, 0><<<blocks, 256, 0, stream>>>(h, src, dst, sum2, deg, nEdges);
  }
  // 6) sum2 -> mean2
  {
    long long total4 = (long long)nNodes * HIDC / 4;
    gs_mean<HIDC><<<(int)((total4 + 255) / 256), 256, 0, stream>>>(
        (float4*)sum2, deg, total4);
  }
  // 7) both heads via WMMA f32
  gs_gemm2<<<dim3(nNodes / 16), dim3(32, 2), 0, stream>>>(
      sum2, h, W2l, W2r, b2, Wdl, Wdr, bd, out, nNodes);
}